// DeformConv_13417477833322
// MI455X (gfx1250) — compile-verified
//
#include <hip/hip_runtime.h>
#include <hip/hip_bf16.h>
#include <math.h>

typedef __attribute__((ext_vector_type(2))) float v2f;
typedef __attribute__((ext_vector_type(8))) float v8f;

#define BDIM 256

constexpr int Bn = 4, Cn = 192, Hn = 64, Wn = 64, KK = 9, Gn = 4;
constexpr int CG = Cn / Gn;   // 48
constexpr int HW = Hn * Wn;   // 4096
constexpr int NOFF = 2 * KK;  // 18

__device__ __forceinline__ v8f wmma_f32_16x16x4(v2f a, v2f b, v8f c) {
#if defined(__HIP_DEVICE_COMPILE__)
  // 8 args: (neg_a, A, neg_b, B, c_mod, C, reuse_a, reuse_b)
  return __builtin_amdgcn_wmma_f32_16x16x4_f32(false, a, false, b, (short)0, c,
                                               false, false);
#else
  (void)a; (void)b;
  return c;
#endif
}

// ---------------------------------------------------------------------------
// Kernel 1: offset-predicting conv  (implicit GEMM, fp32 WMMA)
//   M = 128 positions per block (2 rows), N = 18 (padded to 32), K = 192*9
// ---------------------------------------------------------------------------
__global__ __launch_bounds__(BDIM) void offset_conv_kernel(
    const float* __restrict__ x, const float* __restrict__ w_off,
    const float* __restrict__ b_off, float* __restrict__ offs) {
  __shared__ float Xs[4 * 66 * 18];  // [row 0..3][col 0..65][16 ch, pad 18]
  __shared__ float Ws[9 * 32 * 18];  // [tap][n pad 32][16 ch, pad 18]

  const int tid = threadIdx.x;
  const int b = blockIdx.x >> 5;          // 32 two-row tiles per image
  const int h0 = (blockIdx.x & 31) * 2;
  const int lane = tid & 31;
  const int lhalf = lane >> 4;
  const int lm = lane & 15;
  const int wv = tid >> 5;                // wave id -> M tile = wv*16

  v8f acc[2] = {};

  const int m = wv * 16 + lm;             // A-fragment row for this lane
  const int mrow = m >> 6;
  const int mcol = m & 63;

  for (int cc = 0; cc < Cn / 16; ++cc) {
    // stage padded x slab: rows h0-1..h0+2, cols -1..64, channel-contiguous
    for (int i = tid; i < 4 * 66 * 16; i += BDIM) {
      int r = i / (66 * 16);
      int rem = i - r * (66 * 16);
      int col = rem >> 4;
      int c = rem & 15;
      int y = h0 - 1 + r;
      int xx = col - 1;
      float v = 0.0f;
      if ((unsigned)y < (unsigned)Hn && (unsigned)xx < (unsigned)Wn)
        v = x[((b * Cn + cc * 16 + c) * Hn + y) * Wn + xx];
      Xs[(r * 66 + col) * 18 + c] = v;
    }
    // stage weights, zero-fill N = 18..31
    for (int i = tid; i < 9 * 32 * 16; i += BDIM) {
      int k = i / (32 * 16);
      int rem = i - k * 512;
      int n = rem >> 4;
      int c = rem & 15;
      float v = 0.0f;
      if (n < NOFF) v = w_off[((n * Cn + cc * 16 + c) * KK) + k];
      Ws[(k * 32 + n) * 18 + c] = v;
    }
    __syncthreads();

#pragma unroll
    for (int k = 0; k < KK; ++k) {
      const int ky = k / 3, kx = k % 3;
      const int abase = ((mrow + ky) * 66 + (mcol + kx)) * 18;
#pragma unroll
      for (int kk = 0; kk < 4; ++kk) {
        const int cp = kk * 4 + 2 * lhalf;      // even -> 8B-aligned b64
        v2f a = *(const v2f*)&Xs[abase + cp];
#pragma unroll
        for (int nt = 0; nt < 2; ++nt) {
          v2f bb = *(const v2f*)&Ws[(k * 32 + nt * 16 + lm) * 18 + cp];
          acc[nt] = wmma_f32_16x16x4(a, bb, acc[nt]);
        }
      }
    }
    __syncthreads();
  }

  // epilogue: D layout vgpr r -> (M = r + 8*lhalf, N = lm)
#pragma unroll
  for (int nt = 0; nt < 2; ++nt) {
    int n = nt * 16 + lm;
    if (n < NOFF) {
      float bias = b_off[n];
#pragma unroll
      for (int r = 0; r < 8; ++r) {
        int ml = wv * 16 + r + 8 * lhalf;
        int p = (h0 + (ml >> 6)) * Wn + (ml & 63);
        offs[(b * NOFF + n) * HW + p] = acc[nt][r] + bias;
      }
    }
  }
}

// ---------------------------------------------------------------------------
// Kernel 2: deformable conv (bilinear gather -> LDS -> fp32 WMMA grouped GEMM)
//   per block: (b, g, 128 positions); K = 48 channels x 9 taps
// ---------------------------------------------------------------------------
__global__ __launch_bounds__(BDIM) void deform_conv_kernel(
    const float* __restrict__ x, const float* __restrict__ offs,
    const float* __restrict__ w_def, float* __restrict__ out) {
  __shared__ float Ss[128 * 50];  // sampled tile [m][ci], pad 50 (even)
  __shared__ float Wt[48 * 50];   // tap weights  [n][ci]
  __shared__ float Cw[128 * 4];   // bilinear corner weights (validity folded in)
  __shared__ int   Ca[128 * 4];   // clamped corner addresses within a plane

  const int tid = threadIdx.x;
  const int blk = blockIdx.x;
  const int tile = blk & 31;             // 32 x 128-position tiles per image
  const int g = (blk >> 5) & 3;
  const int b = blk >> 7;
  const int p0 = tile * 128;

  const int lane = tid & 31;
  const int lhalf = lane >> 4;
  const int lm = lane & 15;
  const int wv = tid >> 5;
  const int m0 = wv * 16;

  const float* xg = x + (size_t)(b * Cn + g * CG) * HW;

  v8f acc[3] = {};

  for (int k = 0; k < KK; ++k) {
    const int ky = k / 3, kx = k % 3;
    // --- phase A: bilinear coords per output position ---
    if (tid < 128) {
      const int p = p0 + tid;
      const int h = p >> 6, w = p & 63;
      const float dy = offs[(b * NOFF + 2 * k) * HW + p];
      const float dx = offs[(b * NOFF + 2 * k + 1) * HW + p];
      const float ys = (float)(h - 1 + ky) + dy;
      const float xs = (float)(w - 1 + kx) + dx;
      const float y0f = floorf(ys), x0f = floorf(xs);
      const float fy = ys - y0f, fx = xs - x0f;
      const int y0 = (int)y0f, x0 = (int)x0f;
      const int y1 = y0 + 1, x1 = x0 + 1;
      const float vy0 = ((unsigned)y0 < (unsigned)Hn) ? 1.0f : 0.0f;
      const float vy1 = ((unsigned)y1 < (unsigned)Hn) ? 1.0f : 0.0f;
      const float vx0 = ((unsigned)x0 < (unsigned)Wn) ? 1.0f : 0.0f;
      const float vx1 = ((unsigned)x1 < (unsigned)Wn) ? 1.0f : 0.0f;
      const int y0c = min(max(y0, 0), Hn - 1), y1c = min(max(y1, 0), Hn - 1);
      const int x0c = min(max(x0, 0), Wn - 1), x1c = min(max(x1, 0), Wn - 1);
      Cw[tid * 4 + 0] = (1.f - fy) * (1.f - fx) * vy0 * vx0;
      Cw[tid * 4 + 1] = (1.f - fy) * fx * vy0 * vx1;
      Cw[tid * 4 + 2] = fy * (1.f - fx) * vy1 * vx0;
      Cw[tid * 4 + 3] = fy * fx * vy1 * vx1;
      Ca[tid * 4 + 0] = y0c * Wn + x0c;
      Ca[tid * 4 + 1] = y0c * Wn + x1c;
      Ca[tid * 4 + 2] = y1c * Wn + x0c;
      Ca[tid * 4 + 3] = y1c * Wn + x1c;
    }
    // stage this tap's weights: Wt[n][ci]
    for (int i = tid; i < CG * CG; i += BDIM) {
      int n = i / CG, ci = i - n * CG;
      Wt[n * 50 + ci] = w_def[((g * CG + n) * CG + ci) * KK + k];
    }
    __syncthreads();

    // --- phase B: branch-free bilinear gather into S[m][ci] ---
    for (int i = tid; i < 128 * CG; i += BDIM) {
      int mm = i / CG, ci = i - mm * CG;
      const float* xc = xg + (size_t)ci * HW;
      float v = Cw[mm * 4 + 0] * xc[Ca[mm * 4 + 0]]
              + Cw[mm * 4 + 1] * xc[Ca[mm * 4 + 1]]
              + Cw[mm * 4 + 2] * xc[Ca[mm * 4 + 2]]
              + Cw[mm * 4 + 3] * xc[Ca[mm * 4 + 3]];
      Ss[mm * 50 + ci] = v;
    }
    __syncthreads();

    // --- phase C: WMMA, K = 48 channels as 12 K4 steps, 3 N tiles ---
#pragma unroll
    for (int kk = 0; kk < 12; ++kk) {
      const int cp = kk * 4 + 2 * lhalf;
      v2f a = *(const v2f*)&Ss[(m0 + lm) * 50 + cp];
#pragma unroll
      for (int nt = 0; nt < 3; ++nt) {
        v2f bb = *(const v2f*)&Wt[(nt * 16 + lm) * 50 + cp];
        acc[nt] = wmma_f32_16x16x4(a, bb, acc[nt]);
      }
    }
    __syncthreads();
  }

  // epilogue
#pragma unroll
  for (int nt = 0; nt < 3; ++nt) {
    const int n = g * CG + nt * 16 + lm;
#pragma unroll
    for (int r = 0; r < 8; ++r) {
      const int ml = m0 + r + 8 * lhalf;
      out[(size_t)(b * Cn + n) * HW + p0 + ml] = acc[nt][r];
    }
  }
}

// ---------------------------------------------------------------------------
extern "C" void kernel_launch(void* const* d_in, const int* in_sizes, int n_in,
                              void* d_out, int out_size, void* d_ws,
                              size_t ws_size, hipStream_t stream) {
  (void)in_sizes; (void)n_in; (void)out_size; (void)ws_size;
  const float* x = (const float*)d_in[0];
  const float* w_off = (const float*)d_in[1];
  const float* b_off = (const float*)d_in[2];
  const float* w_def = (const float*)d_in[3];
  float* offs = (float*)d_ws;   // [B][18][H][W] = 1.13 MB scratch
  float* out = (float*)d_out;

  offset_conv_kernel<<<Bn * (Hn / 2), BDIM, 0, stream>>>(x, w_off, b_off, offs);
  deform_conv_kernel<<<Bn * Gn * (HW / 128), BDIM, 0, stream>>>(x, offs, w_def,
                                                                out);
}